// EncoderPrecision_2740189135635
// MI455X (gfx1250) — compile-verified
//
#include <hip/hip_runtime.h>
#include <hip/hip_bf16.h>

// ---------------------------------------------------------------------------
// Problem constants (from reference): B=8, N=1024, Dx=32, H=256, Dz=8
// ---------------------------------------------------------------------------
#define B_   8
#define N_   1024
#define DX   32
#define H_   256
#define DZ   8
#define ROWS (B_ * N_)                 // 8192 rows of x
#define EPS_ 0.001f

#define WAVES_PER_BLOCK 4
#define ROWS_PER_WAVE   16
#define ROWS_PER_BLOCK  (WAVES_PER_BLOCK * ROWS_PER_WAVE)  // 64

typedef float v2f __attribute__((ext_vector_type(2)));
typedef float v4f __attribute__((ext_vector_type(4)));
typedef float v8f __attribute__((ext_vector_type(8)));

// fp32 WMMA: D(16x16) = A(16x4) * B(4x16) + C   -- V_WMMA_F32_16X16X4_F32
__device__ __forceinline__ v8f wmma_f32(v2f a, v2f b, v8f c) {
    return __builtin_amdgcn_wmma_f32_16x16x4_f32(
        /*neg_a=*/false, a, /*neg_b=*/false, b,
        /*c_mod=*/(short)0, c, /*reuse_a=*/false, /*reuse_b=*/false);
}

// ---------------------------------------------------------------------------
// Kernel 1: fused 3-layer MLP, BOTH heads in one launch (gridDim.y == 2):
//   head 0: d = exp(mlp_d(x))   head 1: u = mlp_o(x)
// Each wave32 pushes 16 rows through relu(x@W0+b0) -> relu(h@W1+b1) -> h@W2+b2
// and writes the (B,Dz,N)-layout head output to workspace.
// ---------------------------------------------------------------------------
__global__ __launch_bounds__(WAVES_PER_BLOCK * 32)
void mlp_heads_kernel(const float* __restrict__ x,
                      const float* __restrict__ dW0, const float* __restrict__ db0,
                      const float* __restrict__ dW1, const float* __restrict__ db1,
                      const float* __restrict__ dW2, const float* __restrict__ db2,
                      const float* __restrict__ oW0, const float* __restrict__ ob0,
                      const float* __restrict__ oW1, const float* __restrict__ ob1,
                      const float* __restrict__ oW2, const float* __restrict__ ob2,
                      float* __restrict__ dvec, float* __restrict__ uvec)
{
    __shared__ float lds_h[WAVES_PER_BLOCK][ROWS_PER_WAVE][H_];   // 64 KB

    const int head = blockIdx.y;                 // 0 = diagonal head, 1 = off-diag
    const float* W0 = head ? oW0 : dW0;
    const float* b0 = head ? ob0 : db0;
    const float* W1 = head ? oW1 : dW1;
    const float* b1 = head ? ob1 : db1;
    const float* W2 = head ? oW2 : dW2;
    const float* b2 = head ? ob2 : db2;
    float*       out = head ? uvec : dvec;

    const int wave = threadIdx.x >> 5;
    const int lane = threadIdx.x & 31;
    const int hi   = lane >> 4;           // half-wave selector (K pairs 0,1 vs 2,3)
    const int l    = lane & 15;           // M (for A) / N (for B,C,D)
    const int row0 = blockIdx.x * ROWS_PER_BLOCK + wave * ROWS_PER_WAVE;

    float (&h)[ROWS_PER_WAVE][H_] = lds_h[wave];

    // ---- A fragments of x: 8 K-chunks of 4 (ISA 32-bit 16x4 A layout) ----
    v2f xa[8];
#pragma unroll
    for (int k = 0; k < 8; ++k) {
        const float* p = x + (size_t)(row0 + l) * DX + k * 4 + 2 * hi;
        xa[k] = *(const v2f*)p;
    }

    // ---- Layer 0: (16x32)@(32x256), bias + relu, C-layout -> LDS row-major
#pragma unroll
    for (int nt = 0; nt < 16; ++nt) {
        const int cb = nt * 16 + l;
        v8f acc = {};
#pragma unroll
        for (int k = 0; k < 8; ++k) {
            const int r = k * 4 + 2 * hi;
            v2f bf;
            bf.x = W0[(r + 0) * H_ + cb];
            bf.y = W0[(r + 1) * H_ + cb];
            acc = wmma_f32(xa[k], bf, acc);
        }
        const float bias = b0[cb];
#pragma unroll
        for (int r = 0; r < 8; ++r) {
            float v = acc[r] + bias;
            h[r + 8 * hi][cb] = v > 0.f ? v : 0.f;
        }
    }

    // ---- Layer 1: (16x256)@(256x256). Keep all 16 C tiles live so each
    //      A-fragment read from LDS is reused across the 16 column tiles.
    v8f acc1[16];
#pragma unroll
    for (int nt = 0; nt < 16; ++nt) acc1[nt] = (v8f){};

    for (int k = 0; k < 64; ++k) {                       // K = 256 in chunks of 4
        const int r = k * 4 + 2 * hi;
        const v2f af = *(const v2f*)&h[l][r];            // ds_load_b64
#pragma unroll
        for (int nt = 0; nt < 16; ++nt) {
            v2f bf;
            bf.x = W1[(r + 0) * H_ + nt * 16 + l];
            bf.y = W1[(r + 1) * H_ + nt * 16 + l];
            acc1[nt] = wmma_f32(af, bf, acc1[nt]);
        }
    }
    // bias + relu, write h1 back in place (this wave's region only)
#pragma unroll
    for (int nt = 0; nt < 16; ++nt) {
        const float bias = b1[nt * 16 + l];
#pragma unroll
        for (int r = 0; r < 8; ++r) {
            float v = acc1[nt][r] + bias;
            h[r + 8 * hi][nt * 16 + l] = v > 0.f ? v : 0.f;
        }
    }

    // ---- Layer 2: (16x256)@(256x8), N padded to 16 with zero columns ----
    v8f acc2 = {};
    for (int k = 0; k < 64; ++k) {
        const int r = k * 4 + 2 * hi;
        const v2f af = *(const v2f*)&h[l][r];
        v2f bf;
        bf.x = (l < DZ) ? W2[(r + 0) * DZ + l] : 0.f;
        bf.y = (l < DZ) ? W2[(r + 1) * DZ + l] : 0.f;
        acc2 = wmma_f32(af, bf, acc2);
    }

    // ---- Epilogue: out[(b*DZ + z)*N + n], z = l (valid z < 8) ----
    if (l < DZ) {
        const float bias = b2[l];
#pragma unroll
        for (int r = 0; r < 8; ++r) {
            const int g  = row0 + r + 8 * hi;   // global row in [0, B*N)
            const int bb = g >> 10;             // / N_
            const int n  = g & (N_ - 1);
            float v = acc2[r] + bias;
            if (head == 0) v = __expf(v);
            out[((size_t)bb * DZ + l) * N_ + n] = v;
        }
    }
}

// ---------------------------------------------------------------------------
// Kernel 2: materialize D (diagonal), Bmat (bidiagonal), precision
// (tridiagonal, analytic BtB + eps*I).  One block per output row; each thread
// streams three nontemporal float4 stores (805 MB total -> pure HBM write).
// ---------------------------------------------------------------------------
__global__ __launch_bounds__(256)
void build_outputs_kernel(const float* __restrict__ dvec,
                          const float* __restrict__ uvec,
                          float* __restrict__ out)
{
    const size_t SZ  = (size_t)B_ * DZ * N_ * N_;   // elements per output tensor
    const int    row = blockIdx.x;                  // (b*DZ + z)*N + i
    const int    i   = row & (N_ - 1);

    const float di   = dvec[row];
    const float ui   = uvec[row];
    const float dim1 = (i > 0) ? dvec[row - 1] : 0.f;
    const float uim1 = (i > 0) ? uvec[row - 1] : 0.f;

    const int c0 = threadIdx.x * 4;                 // cols [c0, c0+4)

    float aD[4], aB[4], aP[4];
#pragma unroll
    for (int j = 0; j < 4; ++j) {
        const int col = c0 + j;
        float vD = 0.f, vB = 0.f, vP = 0.f;
        if (col == i) {
            vD = di;
            vB = di;
            vP = di * di + uim1 * uim1 + EPS_;
        } else if (col == i + 1 && i < N_ - 1) {
            vB = ui;                                // superdiagonal (u[:-1])
            vP = di * ui;
        } else if (col == i - 1) {
            vP = dim1 * uim1;                       // subdiagonal of BtB
        }
        aD[j] = vD; aB[j] = vB; aP[j] = vP;
    }

    const size_t base = (size_t)row * N_ + c0;
    v4f pD = { aD[0], aD[1], aD[2], aD[3] };
    v4f pB = { aB[0], aB[1], aB[2], aB[3] };
    v4f pP = { aP[0], aP[1], aP[2], aP[3] };
    __builtin_nontemporal_store(pD, (v4f*)(out + base));
    __builtin_nontemporal_store(pB, (v4f*)(out + SZ + base));
    __builtin_nontemporal_store(pP, (v4f*)(out + 2 * SZ + base));
}

// ---------------------------------------------------------------------------
extern "C" void kernel_launch(void* const* d_in, const int* in_sizes, int n_in,
                              void* d_out, int out_size, void* d_ws, size_t ws_size,
                              hipStream_t stream)
{
    (void)in_sizes; (void)n_in; (void)out_size; (void)ws_size;

    const float* x   = (const float*)d_in[0];
    const float* dW0 = (const float*)d_in[1];
    const float* db0 = (const float*)d_in[2];
    const float* dW1 = (const float*)d_in[3];
    const float* db1 = (const float*)d_in[4];
    const float* dW2 = (const float*)d_in[5];
    const float* db2 = (const float*)d_in[6];
    const float* oW0 = (const float*)d_in[7];
    const float* ob0 = (const float*)d_in[8];
    const float* oW1 = (const float*)d_in[9];
    const float* ob1 = (const float*)d_in[10];
    const float* oW2 = (const float*)d_in[11];
    const float* ob2 = (const float*)d_in[12];

    float* out  = (float*)d_out;
    float* dvec = (float*)d_ws;                    // B*DZ*N floats
    float* uvec = dvec + (size_t)B_ * DZ * N_;     // B*DZ*N floats

    dim3 grid(ROWS / ROWS_PER_BLOCK, 2);           // 128 row-blocks x 2 heads
    mlp_heads_kernel<<<grid, WAVES_PER_BLOCK * 32, 0, stream>>>(
        x, dW0, db0, dW1, db1, dW2, db2,
           oW0, ob0, oW1, ob1, oW2, ob2, dvec, uvec);

    build_outputs_kernel<<<B_ * DZ * N_, 256, 0, stream>>>(dvec, uvec, out);
}